// TrajectoryNet_14723147891169
// MI455X (gfx1250) — compile-verified
//
#include <hip/hip_runtime.h>
#include <math.h>

#define D 16
#define H 64
#define BATCH 32768
#define SPG 16            // samples per workgroup
#define ROWS 272          // 16 forward rows + 16*16 tangent rows = 17 tiles of 16
#define LDW 68            // LDS row stride (mult. of 4 -> 16B-aligned float4 rows)
#define THREADS 256       // 8 wave32s

typedef __attribute__((ext_vector_type(2))) float v2f;
typedef __attribute__((ext_vector_type(4))) float v4f;
typedef __attribute__((ext_vector_type(8))) float v8f;

// One MLP layer applied to the stacked (forward + tangent) row block via
// V_WMMA_F32_16X16X4_F32.  buf holds [ROWS x DIN] on entry, [ROWS x DOUT] on exit.
//
// Tile->wave mapping: each wave owns ONE weight column tile (col = wave % C) and
// TFULL = 16/G row tiles (rt = grp + i*G, always < 16) -> the main k-loop has
// NO guards at all; the 17th row tile (rows 256..271) is handled by the grp==0
// waves in a single wave-uniform epilogue block (one branch per layer).
template<int DIN, int DOUT, bool LAST>
__device__ __forceinline__ void layer_pass(float* __restrict__ buf,
                                           const float* __restrict__ Wl,   // [DIN][DOUT] in LDS
                                           const float* __restrict__ bias, // [DOUT]
                                           const float* __restrict__ gate, // [DOUT] sigmoid(t*g+gb)
                                           const float* __restrict__ thb,  // [DOUT] t*hb
                                           int tid) {
  const int lane = tid & 31;
  const int wave = tid >> 5;
  constexpr int C     = DOUT / 16;   // column tiles (4 or 1)
  constexpr int G     = 8 / C;       // waves sharing one column
  constexpr int TFULL = 16 / G;      // guaranteed-valid row tiles per wave (8 or 2)

  const int col0 = (wave % C) * 16;
  const int grp  = wave / C;         // row-tile group 0..G-1
  const bool epi = (grp == 0);       // this wave also owns row tile 16

  // f32 16x16x4 A layout: lane L -> M = L&15, K pair base = (L>>4)*2
  const int m  = lane & 15;
  const int kb = (lane >> 4) << 1;
  const int n  = lane & 15;

  v8f acc[TFULL];
  v8f accE = {};
  {
    v8f zero8 = {};
#pragma unroll
    for (int i = 0; i < TFULL; ++i) acc[i] = zero8;
  }

  // ---- main GEMM: B fragment hoisted per k-chunk, no guards ----
#pragma unroll
  for (int kc = 0; kc < DIN / 4; ++kc) {
    const int k0 = kc * 4;
    v2f b;
    b[0] = Wl[(k0 + kb)     * DOUT + col0 + n];
    b[1] = Wl[(k0 + kb + 1) * DOUT + col0 + n];
#pragma unroll
    for (int i = 0; i < TFULL; ++i) {
      const int rt = grp + i * G;                   // 0..15: always valid
      const float* arow = buf + (rt * 16 + m) * LDW;
      v2f a;
      a[0] = arow[k0 + kb];
      a[1] = arow[k0 + kb + 1];
      acc[i] = __builtin_amdgcn_wmma_f32_16x16x4_f32(false, a, false, b,
                                                     (short)0, acc[i], false, false);
    }
  }
  // ---- epilogue: row tile 16 (rows 256..271), one wave-uniform branch ----
  if (epi) {
#pragma unroll
    for (int kc = 0; kc < DIN / 4; ++kc) {
      const int k0 = kc * 4;
      v2f b;
      b[0] = Wl[(k0 + kb)     * DOUT + col0 + n];
      b[1] = Wl[(k0 + kb + 1) * DOUT + col0 + n];
      const float* arow = buf + (256 + m) * LDW;
      v2f a;
      a[0] = arow[k0 + kb];
      a[1] = arow[k0 + kb + 1];
      accE = __builtin_amdgcn_wmma_f32_16x16x4_f32(false, a, false, b,
                                                   (short)0, accE, false, false);
    }
  }
  __syncthreads();   // everyone done reading buf before anyone overwrites it

  // ---- scatter D tiles (C/D layout: VGPR r -> M=r / M=r+8) back into buf ----
  {
    const int mh = (lane >> 4) << 3;
    const int cn = col0 + (lane & 15);
#pragma unroll
    for (int i = 0; i < TFULL; ++i) {
      const int row0 = (grp + i * G) * 16 + mh;
#pragma unroll
      for (int r = 0; r < 8; ++r)
        buf[(row0 + r) * LDW + cn] = acc[i][r];
    }
    if (epi) {
#pragma unroll
      for (int r = 0; r < 8; ++r)
        buf[(256 + mh + r) * LDW + cn] = accE[r];
    }
  }
  __syncthreads();

  // ---- forward rows (float4): y = tanh((mm + b)*gate + t*hb) ----
  for (int e = tid; e < 16 * (DOUT / 4); e += THREADS) {
    const int s  = e / (DOUT / 4);
    const int j4 = (e % (DOUT / 4)) * 4;
    v4f mm = *(const v4f*)(buf + s * LDW + j4);
    v4f bb = *(const v4f*)(bias + j4);
    v4f gg = *(const v4f*)(gate + j4);
    v4f tt = *(const v4f*)(thb  + j4);
    v4f a  = (mm + bb) * gg + tt;
    if (!LAST) {
      a[0] = tanhf(a[0]); a[1] = tanhf(a[1]); a[2] = tanhf(a[2]); a[3] = tanhf(a[3]);
    }
    *(v4f*)(buf + s * LDW + j4) = a;
  }
  __syncthreads();

  // ---- tangent rows (float4): u = (mm_u * gate) * (1 - y^2) ----
  for (int e = tid; e < 256 * (DOUT / 4); e += THREADS) {
    const int r  = e / (DOUT / 4);
    const int j4 = (e % (DOUT / 4)) * 4;
    v4f u  = *(const v4f*)(buf + (16 + r) * LDW + j4);
    v4f gg = *(const v4f*)(gate + j4);
    u *= gg;
    if (!LAST) {
      v4f y = *(const v4f*)(buf + (r >> 4) * LDW + j4);  // sample s = r/16
      u *= (1.0f - y * y);
    }
    *(v4f*)(buf + (16 + r) * LDW + j4) = u;
  }
  __syncthreads();
}

#define SMEM_FLOATS (ROWS*LDW + 16*64 + 64*64 + 64*64 + 64*16 + 6*256 + 4*320)

__global__ __launch_bounds__(THREADS, 1)
void trajnet_cnf_kernel(const float* __restrict__ ts, const float* __restrict__ x0,
    const float* W0g, const float* b0g, const float* g0g, const float* gb0g, const float* hb0g,
    const float* W1g, const float* b1g, const float* g1g, const float* gb1g, const float* hb1g,
    const float* W2g, const float* b2g, const float* g2g, const float* gb2g, const float* hb2g,
    const float* W3g, const float* b3g, const float* g3g, const float* gb3g, const float* hb3g,
    float* __restrict__ out)
{
  extern __shared__ float smem[];
  float* buf   = smem;                 // ROWS * LDW
  float* sW0   = buf + ROWS * LDW;     // 16*64
  float* sW1   = sW0 + 16 * 64;        // 64*64
  float* sW2   = sW1 + 64 * 64;        // 64*64
  float* sW3   = sW2 + 64 * 64;        // 64*16
  float* sBias = sW3 + 64 * 16;        // [4][64] (zero-padded for layer 3)
  float* sG    = sBias + 256;
  float* sGB   = sG + 256;
  float* sHB   = sGB + 256;
  float* sGate = sHB + 256;            // sigmoid(t*g+gb) per layer
  float* sThb  = sGate + 256;          // t*hb per layer
  float* zz    = sThb + 256;           // z  state [16][20] (x16, l, v)
  float* zac   = zz + 320;             // RK4 accumulator
  float* ztmp  = zac + 320;            // RK4 stage input
  float* kst   = ztmp + 320;           // odefunc output k

  const int tid  = threadIdx.x;
  const int base = blockIdx.x * SPG;

  // ---- preload weights + per-layer params into LDS ----
  for (int e = tid; e < 16 * 64; e += THREADS) sW0[e] = W0g[e];
  for (int e = tid; e < 64 * 64; e += THREADS) sW1[e] = W1g[e];
  for (int e = tid; e < 64 * 64; e += THREADS) sW2[e] = W2g[e];
  for (int e = tid; e < 64 * 16; e += THREADS) sW3[e] = W3g[e];
  {
    const int l = tid >> 6, j = tid & 63;
    const int dl = (l == 3) ? 16 : 64;
    const float* bp  = (l==0)?b0g :(l==1)?b1g :(l==2)?b2g :b3g;
    const float* gp  = (l==0)?g0g :(l==1)?g1g :(l==2)?g2g :g3g;
    const float* gbp = (l==0)?gb0g:(l==1)?gb1g:(l==2)?gb2g:gb3g;
    const float* hbp = (l==0)?hb0g:(l==1)?hb1g:(l==2)?hb2g:hb3g;
    float bv = 0.f, gv = 0.f, gbv = 0.f, hbv = 0.f;
    if (j < dl) { bv = bp[j]; gv = gp[j]; gbv = gbp[j]; hbv = hbp[j]; }
    sBias[tid] = bv; sG[tid] = gv; sGB[tid] = gbv; sHB[tid] = hbv;
  }

  // ---- init z = pad(x0), emit xs[0] = x0 ----
  for (int e = tid; e < SPG * 18; e += THREADS) {
    const int s = e / 18, c = e % 18;
    zz[s * 20 + c] = (c < 16) ? x0[(base + s) * 16 + c] : 0.0f;
  }
  {
    const int s = tid >> 4, j = tid & 15;          // 256 threads == 16*16
    out[(base + s) * 16 + j] = x0[(base + s) * 16 + j];
  }
  __syncthreads();

  // odefunc: reads ztmp.x, writes k = (dx[16], dl, dv) into kst
  auto odefunc = [&](float t) {
    { // gates depend only on t: one element per thread ([4][64])
      const float xg = t * sG[tid] + sGB[tid];
      sGate[tid] = 1.0f / (1.0f + expf(-xg));
      sThb[tid]  = t * sHB[tid];
    }
    // init row block (float4): rows 0..15 = x, rows 16..271 = identity tangents
    for (int e = tid; e < ROWS * 4; e += THREADS) {
      const int r  = e >> 2;
      const int j4 = (e & 3) * 4;
      v4f v;
      if (r < 16) {
        v = *(const v4f*)(ztmp + r * 20 + j4);
      } else {
        const int i = (r - 16) & 15;
        v[0] = (i == j4    ) ? 1.0f : 0.0f;
        v[1] = (i == j4 + 1) ? 1.0f : 0.0f;
        v[2] = (i == j4 + 2) ? 1.0f : 0.0f;
        v[3] = (i == j4 + 3) ? 1.0f : 0.0f;
      }
      *(v4f*)(buf + r * LDW + j4) = v;
    }
    __syncthreads();

    layer_pass<16, 64, false>(buf, sW0, sBias +   0, sGate +   0, sThb +   0, tid);
    layer_pass<64, 64, false>(buf, sW1, sBias +  64, sGate +  64, sThb +  64, tid);
    layer_pass<64, 64, false>(buf, sW2, sBias + 128, sGate + 128, sThb + 128, tid);
    layer_pass<64, 16, true >(buf, sW3, sBias + 192, sGate + 192, sThb + 192, tid);

    if (tid < SPG) {   // per-sample: dx, trace(J), 0.5*||dx||^2
      const int s = tid;
      float dv = 0.0f;
#pragma unroll
      for (int j = 0; j < 16; ++j) {
        const float d = buf[s * LDW + j];
        kst[s * 20 + j] = d;
        dv += d * d;
      }
      float dl = 0.0f;
#pragma unroll
      for (int i = 0; i < 16; ++i) dl += buf[(16 + s * 16 + i) * LDW + i];
      kst[s * 20 + 16] = dl;
      kst[s * 20 + 17] = 0.5f * dv;
    }
    __syncthreads();
  };

  // ---- RK4 over ts intervals ----
#pragma unroll 1
  for (int step = 0; step < 2; ++step) {
    const float t0 = ts[step];
    const float t1 = ts[step + 1];
    const float hh = t1 - t0;

    for (int e = tid; e < SPG * 18; e += THREADS) {
      const int s = e / 18, c = e % 18;
      const float v = zz[s * 20 + c];
      zac[s * 20 + c] = v; ztmp[s * 20 + c] = v;
    }
    __syncthreads();

#pragma unroll 1
    for (int stage = 0; stage < 4; ++stage) {
      const float tst = (stage == 0) ? t0 : ((stage == 3) ? t1 : t0 + 0.5f * hh);
      odefunc(tst);
      const float wa = ((stage == 0) || (stage == 3)) ? hh * (1.0f / 6.0f)
                                                      : hh * (1.0f / 3.0f);
      const float wn = (stage < 2) ? 0.5f * hh : hh;
      for (int e = tid; e < SPG * 18; e += THREADS) {
        const int s = e / 18, c = e % 18;
        const float k = kst[s * 20 + c];
        zac[s * 20 + c] += wa * k;
        if (stage < 3) ztmp[s * 20 + c] = zz[s * 20 + c] + wn * k;
      }
      __syncthreads();
    }

    for (int e = tid; e < SPG * 18; e += THREADS) {
      const int s = e / 18, c = e % 18;
      zz[s * 20 + c] = zac[s * 20 + c];
    }
    __syncthreads();
    {
      const int s = tid >> 4, j = tid & 15;
      out[((size_t)(step + 1) * BATCH + base + s) * 16 + j] = zz[s * 20 + j];
    }
    __syncthreads();
  }

  // ---- final scalars: log_det, |v|, zeros ----
  if (tid < SPG) {
    const int gs = base + tid;
    out[(size_t)3 * BATCH * 16 + gs]             = zz[tid * 20 + 16];
    out[(size_t)3 * BATCH * 16 + BATCH + gs]     = fabsf(zz[tid * 20 + 17]);
    out[(size_t)3 * BATCH * 16 + 2 * BATCH + gs] = 0.0f;
  }
}

extern "C" void kernel_launch(void* const* d_in, const int* in_sizes, int n_in,
                              void* d_out, int out_size, void* d_ws, size_t ws_size,
                              hipStream_t stream) {
  (void)in_sizes; (void)n_in; (void)out_size; (void)d_ws; (void)ws_size;
  const float* p[22];
  for (int i = 0; i < 22; ++i) p[i] = (const float*)d_in[i];
  const size_t shmem = (size_t)SMEM_FLOATS * sizeof(float);   // ~123 KB dynamic LDS
  dim3 grid(BATCH / SPG), block(THREADS);
  hipLaunchKernelGGL(trajnet_cnf_kernel, grid, block, shmem, stream,
      p[0],  p[1],
      p[2],  p[3],  p[4],  p[5],  p[6],
      p[7],  p[8],  p[9],  p[10], p[11],
      p[12], p[13], p[14], p[15], p[16],
      p[17], p[18], p[19], p[20], p[21],
      (float*)d_out);
}